// GlobalAttention_4715874091389
// MI455X (gfx1250) — compile-verified
//
#include <hip/hip_runtime.h>
#include <hip/hip_bf16.h>
#include <stdint.h>

#define FDIM 256
#define NWAVES 8
#define EPSV 1e-16f

typedef __attribute__((ext_vector_type(2))) float v2f;
typedef __attribute__((ext_vector_type(8))) float v8f;

// ---------------------------------------------------------------------------
// Kernel 1: per-segment fused gate GEMV + segment softmax + weighted row-sum.
//   S[g,:]   = sum_i alpha_i * x[i,:]      (i in segment g)
//   asum[g]  = sum_i alpha_i  ( = s/(s+EPS) )
// batch is sorted int64 -> each block binary-searches its row range.
// ---------------------------------------------------------------------------
__global__ void __launch_bounds__(256)
ga_segment_kernel(const float* __restrict__ x,
                  const long long* __restrict__ batch,
                  const float* __restrict__ Wg,
                  const float* __restrict__ bg,
                  float* __restrict__ S,
                  float* __restrict__ asum,
                  int N)
{
    __shared__ float sWg[FDIM];
    __shared__ float sAcc[NWAVES][FDIM];
    __shared__ float sRed[NWAVES];
    __shared__ int   sBounds[2];
    __shared__ float sBcast;

    const int g    = blockIdx.x;
    const int tid  = threadIdx.x;
    const int lane = tid & 31;          // wave32
    const int wave = tid >> 5;

    sWg[tid] = Wg[tid];                 // blockDim.x == FDIM
    if (tid == 0) {
        // lower_bound(batch, g) and lower_bound(batch, g+1)
        long long tg = (long long)g;
        int lo = 0, hi = N;
        while (lo < hi) { int mid = (lo + hi) >> 1; if (batch[mid] < tg) lo = mid + 1; else hi = mid; }
        sBounds[0] = lo;
        int lo2 = lo, hi2 = N; long long tg1 = tg + 1;
        while (lo2 < hi2) { int mid = (lo2 + hi2) >> 1; if (batch[mid] < tg1) lo2 = mid + 1; else hi2 = mid; }
        sBounds[1] = lo2;
    }
    __syncthreads();
    const int rlo = sBounds[0], rhi = sBounds[1];
    const float bgv = bg[0];

    // ---- Phase 1: segment max of gate = x@Wg + bg ----
    float wmax = -3.402823466e38f;
    for (int r = rlo + wave; r < rhi; r += NWAVES) {
        const float* xr = x + (size_t)r * FDIM;
        float p = 0.f;
#pragma unroll
        for (int k = 0; k < 8; ++k) p += xr[lane + 32 * k] * sWg[lane + 32 * k];
#pragma unroll
        for (int off = 16; off >= 1; off >>= 1) p += __shfl_xor(p, off, 32);
        wmax = fmaxf(wmax, p + bgv);
    }
    if (lane == 0) sRed[wave] = wmax;
    __syncthreads();
    if (tid == 0) {
        float m = sRed[0];
        for (int w = 1; w < NWAVES; ++w) m = fmaxf(m, sRed[w]);
        sBcast = m;
    }
    __syncthreads();
    const float m = sBcast;
    __syncthreads();   // everyone has m before sBcast is reused below

    // ---- Phase 2: e = exp(gate - m); acc += e*x; s += e ----
    // Second read of this segment's x rows (~200KB) hits L2/WGP$.
    float acc[8];
#pragma unroll
    for (int k = 0; k < 8; ++k) acc[k] = 0.f;
    float s = 0.f;
    for (int r = rlo + wave; r < rhi; r += NWAVES) {
        const float* xr = x + (size_t)r * FDIM;
        float xv[8];
        float p = 0.f;
#pragma unroll
        for (int k = 0; k < 8; ++k) { xv[k] = xr[lane + 32 * k]; p += xv[k] * sWg[lane + 32 * k]; }
#pragma unroll
        for (int off = 16; off >= 1; off >>= 1) p += __shfl_xor(p, off, 32);
        const float e = __expf(p + bgv - m);   // wave-uniform
        s += e;
#pragma unroll
        for (int k = 0; k < 8; ++k) acc[k] = fmaf(e, xv[k], acc[k]);
    }
#pragma unroll
    for (int k = 0; k < 8; ++k) sAcc[wave][lane + 32 * k] = acc[k];
    if (lane == 0) sRed[wave] = s;
    __syncthreads();
    if (tid == 0) {
        float st = 0.f;
        for (int w = 0; w < NWAVES; ++w) st += sRed[w];
        sBcast = st;
        asum[g] = st / (st + EPSV);
    }
    __syncthreads();
    const float stot = sBcast;
    float v = 0.f;
#pragma unroll
    for (int w = 0; w < NWAVES; ++w) v += sAcc[w][tid];
    S[(size_t)g * FDIM + tid] = v / (stot + EPSV);
}

// ---------------------------------------------------------------------------
// Kernel 2: out = S @ Wn + asum (x) bn   via V_WMMA_F32_16X16X4_F32.
// One wave per 16x16 output tile; K loop of 64 steps (K=4 each).
// A layout (f32 16x4): lane l -> M = l&15, K = {kb, kb+1}, kb = (l>>4)*2.
// B layout (f32 4x16): lane l -> N = l&15, K rows {kb, kb+1} in v0/v1.
// C/D layout: lane l, vgpr r -> M = r + (l>>4)*8, N = l&15.
// ---------------------------------------------------------------------------
__global__ void __launch_bounds__(256)
ga_gemm_wmma_kernel(const float* __restrict__ S,
                    const float* __restrict__ Wn,
                    const float* __restrict__ bn,
                    const float* __restrict__ asum,
                    float* __restrict__ out,
                    int G)
{
    const int tid      = threadIdx.x;
    const int lane     = tid & 31;
    const int waveGbl  = blockIdx.x * (blockDim.x >> 5) + (tid >> 5);
    const int tilesN   = FDIM / 16;             // 16
    const int tileM    = waveGbl / tilesN;
    const int tileN    = waveGbl % tilesN;
    if (tileM * 16 >= G) return;                // wave-uniform exit: EXEC stays all-1 for WMMA

    const int halfSel = lane >> 4;              // 0 | 1
    const int idx16   = lane & 15;
    const int koff    = halfSel * 2;            // A/B K sub-offset

    v8f c = {};
    const float* Srow = S + (size_t)(tileM * 16 + idx16) * FDIM;   // A: row M = idx16
    const float* Wcol = Wn + (size_t)(tileN * 16 + idx16);         // B: col N = idx16

    for (int k = 0; k < FDIM; k += 4) {
        v2f a, b;
        a.x = Srow[k + koff];
        a.y = Srow[k + koff + 1];
        const float* wp = Wcol + (size_t)(k + koff) * FDIM;
        b.x = wp[0];
        b.y = wp[FDIM];
        // 8 args: (neg_a, A, neg_b, B, c_mod, C, reuse_a, reuse_b)
        c = __builtin_amdgcn_wmma_f32_16x16x4_f32(
                false, a, false, b, (short)0, c, false, false);
    }

    const int nIdx = tileN * 16 + idx16;
    const float bnv = bn[nIdx];
#pragma unroll
    for (int r = 0; r < 8; ++r) {
        const int mIdx = tileM * 16 + r + halfSel * 8;
        out[(size_t)mIdx * FDIM + nIdx] = c[r] + asum[mIdx] * bnv;
    }
}

// ---------------------------------------------------------------------------
// Launch: derive N from in_sizes, G from out_size (host-side; `size` scalar
// lives on device and cannot be read during graph capture).
// ---------------------------------------------------------------------------
extern "C" void kernel_launch(void* const* d_in, const int* in_sizes, int n_in,
                              void* d_out, int out_size, void* d_ws, size_t ws_size,
                              hipStream_t stream)
{
    const float*     x     = (const float*)d_in[0];
    const long long* batch = (const long long*)d_in[1];
    // d_in[2] = size (unused on host; G derived from out_size)
    const float*     Wg    = (const float*)d_in[3];
    const float*     bg    = (const float*)d_in[4];
    const float*     Wn    = (const float*)d_in[5];
    const float*     bn    = (const float*)d_in[6];
    float*           out   = (float*)d_out;

    const int N = in_sizes[0] / FDIM;
    const int G = out_size / FDIM;

    float* S    = (float*)d_ws;                   // [G, FDIM]
    float* asum = S + (size_t)G * FDIM;           // [G]

    // Pass 1: segment softmax + weighted segment sums (HBM-bound, ~205 MB)
    ga_segment_kernel<<<G, 256, 0, stream>>>(x, batch, Wg, bg, S, asum, N);

    // Pass 2: tiny WMMA GEMM  out = S @ Wn + asum (x) bn
    const int totalWaves = (G / 16) * (FDIM / 16);
    const int blocks     = (totalWaves + NWAVES - 1) / NWAVES;
    ga_gemm_wmma_kernel<<<blocks, 256, 0, stream>>>(S, Wn, bn, asum, out, G);
}